// Conv2DLayer_16011638080159
// MI455X (gfx1250) — compile-verified
//
#include <hip/hip_runtime.h>

typedef __attribute__((ext_vector_type(2))) float v2f;
typedef __attribute__((ext_vector_type(8))) float v8f;

#define CDIM 512
#define KDIM 9

// Kernel 1: fold the 9 weight rows and the "* C" factor into one 512-float vector.
__global__ void Conv2DLayer_wsum_kernel(const float* __restrict__ weight,
                                        float* __restrict__ ws) {
    int c = threadIdx.x;            // launched with 512 threads
    float s = 0.0f;
#pragma unroll
    for (int k = 0; k < KDIM; ++k) s += weight[k * CDIM + c];
    ws[c] = (float)CDIM * s;
}

// Kernel 2: out = X @ ws via V_WMMA_F32_16X16X4_F32.
// Each wave owns a 16-row tile; 128 WMMAs sweep C=512 in K=4 steps.
// A fragment (16x4 f32): lanes 0-15 hold row M=lane, K={0,1}; lanes 16-31 hold
// row M=lane-16, K={2,3}  -> per-lane v2f at x[row] + c0 + 2*hi.
// B fragment (4x16 f32): every column n gets ws[c0+k]; per-lane v2f at
// ws + c0 + 2*hi (identical across all 16 columns).
// All 16 columns of D are then identical, so lane 0 (M=0..7 in VGPR0..7) and
// lane 16 (M=8..15) hold the whole 16-row output tile.
__global__ void __launch_bounds__(256)
Conv2DLayer_gemv_wmma_kernel(const float* __restrict__ x,
                             const float* __restrict__ ws,
                             float* __restrict__ out) {
    __shared__ v2f lws2[CDIM / 2];
    // 256 threads stage 512 floats: one aligned b64 each, no loop.
    lws2[threadIdx.x] = ((const v2f*)ws)[threadIdx.x];
    __syncthreads();

    const int lane = threadIdx.x & 31;
    const int wv   = threadIdx.x >> 5;      // wave id in block: 0..7
    const int m    = lane & 15;             // row within tile
    const int hi   = lane >> 4;             // K half: 0 -> K={0,1} ; 1 -> K={2,3}

    const long tile = (long)blockIdx.x * 8 + wv;
    const float* xp = x + tile * 16 * CDIM + (long)m * CDIM + 2 * hi;
    const float* wp = (const float*)lws2 + 2 * hi;

    v8f acc0 = {};
    v8f acc1 = {};
#pragma unroll 8
    for (int c0 = 0; c0 < CDIM; c0 += 8) {
        // X is streamed exactly once (1.07 GB) -> non-temporal loads keep L2
        // for reusable data; ws lives in LDS (broadcast reads, no conflicts).
        v2f a0 = __builtin_nontemporal_load((const v2f*)(xp + c0));
        v2f b0 = *(const v2f*)(wp + c0);
        v2f a1 = __builtin_nontemporal_load((const v2f*)(xp + c0 + 4));
        v2f b1 = *(const v2f*)(wp + c0 + 4);
        // 8 args: (neg_a, A, neg_b, B, c_mod, C, reuse_a, reuse_b)
        acc0 = __builtin_amdgcn_wmma_f32_16x16x4_f32(false, a0, false, b0,
                                                     (short)0, acc0, false, false);
        acc1 = __builtin_amdgcn_wmma_f32_16x16x4_f32(false, a1, false, b1,
                                                     (short)0, acc1, false, false);
    }
    v8f acc = acc0 + acc1;

    // Column 0 lives in lanes 0 (M=0..7) and 16 (M=8..15); 2 x 32B contiguous
    // non-temporal stores cover the tile (out is written once, never re-read).
    if (m == 0) {
        float* o = out + tile * 16 + hi * 8;
        __builtin_nontemporal_store(acc, (v8f*)o);
    }
}

extern "C" void kernel_launch(void* const* d_in, const int* in_sizes, int n_in,
                              void* d_out, int out_size, void* d_ws, size_t ws_size,
                              hipStream_t stream) {
    const float* x      = (const float*)d_in[0];   // [B, 512]
    const float* weight = (const float*)d_in[1];   // [9, 512]
    float*       out    = (float*)d_out;           // [B]
    float*       ws     = (float*)d_ws;            // 512 floats scratch

    const int Btotal = in_sizes[0] / CDIM;         // 524288
    const int tiles  = Btotal / 16;                // 32768
    const int blocks = tiles / 8;                  // 4096 (8 waves per block)

    Conv2DLayer_wsum_kernel<<<1, CDIM, 0, stream>>>(weight, ws);
    Conv2DLayer_gemv_wmma_kernel<<<blocks, 256, 0, stream>>>(x, ws, out);
}